// Mamba_73830487818391
// MI455X (gfx1250) — compile-verified
//
#include <hip/hip_runtime.h>
#include <hip/hip_bf16.h>

typedef __attribute__((ext_vector_type(2))) float v2f;
typedef __attribute__((ext_vector_type(8))) float v8f;
typedef __attribute__((ext_vector_type(4))) unsigned int v4u;
typedef __attribute__((ext_vector_type(4))) int v4i;
typedef __attribute__((ext_vector_type(8))) int v8i;

#define D_MODEL 1024
#define D_INNER 2048
#define D_OUT   1024
#define SEQLEN  8192

// GEMM-1 staging geometry
#define BM   64      // block M tile
#define BN   128     // block N tile
#define KB   32      // K per stage
#define LSTR 36      // padded LDS row stride (floats): 32 data + 4 pad -> conflict-free ds_load_b64

#if defined(__HIP_DEVICE_COMPILE__) && \
    __has_builtin(__builtin_amdgcn_tensor_load_to_lds) && \
    __has_builtin(__builtin_amdgcn_s_wait_tensorcnt)
#define HAVE_TDM 1
#else
#define HAVE_TDM 0
#endif

// D = A(16x4) * B(4x16) + C on the fp32 matrix pipe
__device__ __forceinline__ v8f wmma_f32(v2f a, v2f b, v8f c) {
    return __builtin_amdgcn_wmma_f32_16x16x4_f32(
        /*neg_a=*/false, a, /*neg_b=*/false, b,
        /*c_mod=*/(short)0, c, /*reuse_a=*/false, /*reuse_b=*/false);
}

#if HAVE_TDM
__device__ __forceinline__ unsigned lds_off(const void* p) {
    // generic LDS pointer: low 32 bits are the LDS byte offset
    return (unsigned)(uintptr_t)p;
}

// 2-D TDM tile load: tile_h rows x KB floats, row stride `row_stride` floats,
// LDS padding 4 dwords every 32 dwords -> effective LDS row stride LSTR floats.
__device__ __forceinline__ void tdm_load_2d(
    const float* gsrc, unsigned lds_byte,
    unsigned tensor_d0, unsigned tensor_d1, unsigned row_stride, unsigned tile_h)
{
    unsigned long long ga = (unsigned long long)(uintptr_t)gsrc;
    v4u g0;
    g0.x = 1u;                                           // count=1 (valid user D#)
    g0.y = lds_byte;                                     // lds_addr
    g0.z = (unsigned)(ga & 0xffffffffull);               // global_addr[31:0]
    g0.w = (unsigned)((ga >> 32) & 0x01ffffffull)        // global_addr[56:32]
         | (2u << 30);                                   // type = 2 (image/tensor)
    v8i g1;
    g1[0] = (int)((2u << 16)      // data_size = 4B
                | (1u << 20)      // pad_enable
                | (4u << 22)      // pad_interval: every 32 dwords
                | (3u << 25));    // pad_amount: 4 dwords
    g1[1] = (int)(tensor_d0 << 16);                          // tensor_dim0[15:0]
    g1[2] = (int)((tensor_d0 >> 16) | (tensor_d1 << 16));    // dim0 hi | dim1 lo
    g1[3] = (int)((tensor_d1 >> 16) | ((unsigned)KB << 16)); // dim1 hi | tile_dim0
    g1[4] = (int)(tile_h & 0xffffu);                         // tile_dim1 (tile_dim2=0)
    g1[5] = (int)row_stride;                                 // tensor_dim0_stride lo
    g1[6] = 0;                                               // stride hi | dim1_stride lo
    g1[7] = 0;
    v4i z4 = {0, 0, 0, 0};
#if __clang_major__ >= 23
    v8i z8 = {0, 0, 0, 0, 0, 0, 0, 0};
    __builtin_amdgcn_tensor_load_to_lds(g0, g1, z4, z4, z8, 0);
#else
    __builtin_amdgcn_tensor_load_to_lds(g0, g1, z4, z4, 0);
#endif
}
#endif // HAVE_TDM

// Issue one K-stage of operand staging into LDS (TDM if available, else
// cooperative coalesced b128 copy into the same padded layout).
__device__ __forceinline__ void issue_stage(
    const float* __restrict__ A, const float* __restrict__ W,
    float* sA, float* sB, int m0, int n0, int k, int M, int tid)
{
#if HAVE_TDM
    if ((tid >> 5) == 0) {   // one wave issues the DMAs for the workgroup
        tdm_load_2d(A + (size_t)m0 * D_MODEL + k, lds_off(sA),
                    D_MODEL, (unsigned)M, D_MODEL, BM);
        tdm_load_2d(W + (size_t)n0 * D_MODEL + k, lds_off(sB),
                    D_MODEL, D_INNER, D_MODEL, BN);
    }
#else
#pragma unroll
    for (int i = 0; i < (BM * KB / 4) / 256; ++i) {       // 2 iters
        int idx = tid + i * 256;
        int row = idx >> 3, c4 = (idx & 7) * 4;
        *(float4*)&sA[row * LSTR + c4] =
            *(const float4*)&A[(size_t)(m0 + row) * D_MODEL + k + c4];
    }
#pragma unroll
    for (int i = 0; i < (BN * KB / 4) / 256; ++i) {       // 4 iters
        int idx = tid + i * 256;
        int row = idx >> 3, c4 = (idx & 7) * 4;
        *(float4*)&sB[row * LSTR + c4] =
            *(const float4*)&W[(size_t)(n0 + row) * D_MODEL + k + c4];
    }
#endif
}

// One K-stage of WMMA math out of LDS (ds_load_b64 fragments, conflict-free).
__device__ __forceinline__ void compute_stage(
    const float* cA, const float* cB, v8f (&acc)[2][2],
    int wm, int wn, int r, int h)
{
#pragma unroll
    for (int kk = 0; kk < KB; kk += 4) {
        v2f a[2], b[2];
#pragma unroll
        for (int mt = 0; mt < 2; ++mt)
            a[mt] = *(const v2f*)&cA[(wm * 32 + mt * 16 + r) * LSTR + kk + 2 * h];
#pragma unroll
        for (int nt = 0; nt < 2; ++nt)
            b[nt] = *(const v2f*)&cB[(wn * 32 + nt * 16 + r) * LSTR + kk + 2 * h];
#pragma unroll
        for (int mt = 0; mt < 2; ++mt)
#pragma unroll
            for (int nt = 0; nt < 2; ++nt)
                acc[mt][nt] = wmma_f32(a[mt], b[nt], acc[mt][nt]);
    }
}

// ---------------------------------------------------------------------------
// Kernel 1: X[m, n] = sum_d hidden[m, d] * in_proj_w[n, d]   (n < 2048 half)
// TDM double-buffered LDS staging; 8 waves as 2(M) x 4(N); wave tile 32x32.
// ---------------------------------------------------------------------------
__global__ __launch_bounds__(256) void mamba_inproj_wmma(
    const float* __restrict__ A,     // [M x 1024] hidden (b,l flattened)
    const float* __restrict__ W,     // [4096 x 1024] in_proj_w (rows 0..2047 used)
    float* __restrict__ X,           // [M x 2048] workspace
    int M)
{
    __shared__ float sA[2][BM * LSTR];   // 18 KB
    __shared__ float sB[2][BN * LSTR];   // 36 KB

    const int tid  = threadIdx.x;
    const int lane = tid & 31;
    const int wave = tid >> 5;
    const int wm   = wave & 1;
    const int wn   = wave >> 1;
    const int m0   = blockIdx.x * BM;
    const int n0   = blockIdx.y * BN;
    const int r    = lane & 15;
    const int h    = lane >> 4;

    v8f acc[2][2];
#pragma unroll
    for (int mt = 0; mt < 2; ++mt)
#pragma unroll
        for (int nt = 0; nt < 2; ++nt)
            acc[mt][nt] = {};

    const int S = D_MODEL / KB;          // 32 stages

    issue_stage(A, W, sA[0], sB[0], m0, n0, 0, M, tid);

    for (int s = 0; s < S - 1; ++s) {
        issue_stage(A, W, sA[(s + 1) & 1], sB[(s + 1) & 1],
                    m0, n0, (s + 1) * KB, M, tid);
#if HAVE_TDM
        if (wave == 0)
            __builtin_amdgcn_s_wait_tensorcnt((short)2);  // stage s landed
#endif
        __syncthreads();
        compute_stage(sA[s & 1], sB[s & 1], acc, wm, wn, r, h);
        __syncthreads();
    }
#if HAVE_TDM
    if (wave == 0)
        __builtin_amdgcn_s_wait_tensorcnt((short)0);
#endif
    __syncthreads();
    compute_stage(sA[(S - 1) & 1], sB[(S - 1) & 1], acc, wm, wn, r, h);

    // C/D layout: element v -> row (v + 8h), col r
#pragma unroll
    for (int mt = 0; mt < 2; ++mt)
#pragma unroll
        for (int nt = 0; nt < 2; ++nt)
#pragma unroll
            for (int v = 0; v < 8; ++v)
                X[(size_t)(m0 + wm * 32 + mt * 16 + v + 8 * h) * D_INNER +
                  (n0 + wn * 32 + nt * 16 + r)] = acc[mt][nt][v];
}

// ---------------------------------------------------------------------------
// Kernel 2: out[m, n] = sum_c silu(conv(X)[m, c]) * out_proj_w[n, c]
// conv+bias+SiLU fused into the A-fragment path (branchless causal masking,
// EXEC stays all-ones for WMMA).
// ---------------------------------------------------------------------------
__device__ __forceinline__ v2f conv_silu_frag(
    const float* __restrict__ X,
    const float* __restrict__ cw,   // [2048*4] depthwise taps
    const float* __restrict__ cb,   // [2048]
    int m, int c)                   // channels c, c+1
{
    const int l = m & (SEQLEN - 1);          // position within batch sequence
    float s0 = cb[c];
    float s1 = cb[c + 1];
#pragma unroll
    for (int j = 0; j < 4; ++j) {
        const int dl   = j - 3;              // taps cover l-3 .. l
        const bool ok  = (l + dl) >= 0;      // causal zero-pad (per batch)
        const float mk = ok ? 1.0f : 0.0f;
        const int mm   = ok ? (m + dl) : m;  // clamped, always in-bounds
        const float* p = X + (size_t)mm * D_INNER + c;
        s0 += cw[c * 4 + j]       * (mk * p[0]);
        s1 += cw[(c + 1) * 4 + j] * (mk * p[1]);
    }
    v2f y;
    y.x = s0 / (1.0f + __expf(-s0));
    y.y = s1 / (1.0f + __expf(-s1));
    return y;
}

__global__ __launch_bounds__(256) void mamba_conv_outproj_wmma(
    const float* __restrict__ X,     // [M x 2048]
    const float* __restrict__ cw,    // [2048 x 1 x 4]
    const float* __restrict__ cb,    // [2048]
    const float* __restrict__ W,     // [1024 x 2048] out_proj_w
    float* __restrict__ Out,         // [M x 1024]
    int M)
{
    const int K = D_INNER;
    const int N = D_OUT;

    const int lane = threadIdx.x & 31;
    const int wave = threadIdx.x >> 5;
    const int wm   = wave & 1;
    const int wn   = wave >> 1;
    const int m0   = blockIdx.x * 64  + wm * 32;
    const int n0   = blockIdx.y * 256 + wn * 64;
    const int r    = lane & 15;
    const int h    = lane >> 4;

    v8f acc[2][4];
#pragma unroll
    for (int mt = 0; mt < 2; ++mt)
#pragma unroll
        for (int nt = 0; nt < 4; ++nt)
            acc[mt][nt] = {};

    const int mA0 = m0 + r;
    const int mA1 = m0 + 16 + r;
    const float* bBase[4];
#pragma unroll
    for (int nt = 0; nt < 4; ++nt)
        bBase[nt] = W + (size_t)(n0 + nt * 16 + r) * K + 2 * h;

#pragma unroll 2
    for (int k = 0; k < K; k += 4) {
        const int c = k + 2 * h;             // this lane's two channels
        v2f a0 = conv_silu_frag(X, cw, cb, mA0, c);
        v2f a1 = conv_silu_frag(X, cw, cb, mA1, c);
        v2f b[4];
#pragma unroll
        for (int nt = 0; nt < 4; ++nt)
            b[nt] = *(const v2f*)(bBase[nt] + k);
#pragma unroll
        for (int nt = 0; nt < 4; ++nt) {
            acc[0][nt] = wmma_f32(a0, b[nt], acc[0][nt]);
            acc[1][nt] = wmma_f32(a1, b[nt], acc[1][nt]);
        }
    }

#pragma unroll
    for (int mt = 0; mt < 2; ++mt)
#pragma unroll
        for (int nt = 0; nt < 4; ++nt)
#pragma unroll
            for (int v = 0; v < 8; ++v)
                Out[(size_t)(m0 + mt * 16 + v + 8 * h) * N + (n0 + nt * 16 + r)] =
                    acc[mt][nt][v];
}

// ---------------------------------------------------------------------------
// Launcher. Inputs (setup_inputs order):
//  0 hidden_states (2,8192,1024) f32   1 in_proj_w (4096,1024)
//  2 conv_w (2048,1,4)                 3 conv_b (2048)
//  4 x_proj_w  5 dt_proj_w  6 dt_proj_b  7 A_log  8 D     (all dead code)
//  9 out_proj_w (1024,2048)
// Output: (2,8192,1024) f32. Workspace: X = 16384*2048 f32 = 128 MB.
// ---------------------------------------------------------------------------
extern "C" void kernel_launch(void* const* d_in, const int* in_sizes, int n_in,
                              void* d_out, int out_size, void* d_ws, size_t ws_size,
                              hipStream_t stream) {
    const float* hidden   = (const float*)d_in[0];
    const float* in_proj  = (const float*)d_in[1];
    const float* conv_w   = (const float*)d_in[2];
    const float* conv_b   = (const float*)d_in[3];
    const float* out_proj = (const float*)d_in[9];

    float* X   = (float*)d_ws;
    float* out = (float*)d_out;

    const int M = in_sizes[0] / D_MODEL;   // 16384 = batch * seqlen

    dim3 blk(256);
    dim3 g1(M / BM, D_INNER / BN);
    mamba_inproj_wmma<<<g1, blk, 0, stream>>>(hidden, in_proj, X, M);

    dim3 g2(M / 64, D_OUT / 256);
    mamba_conv_outproj_wmma<<<g2, blk, 0, stream>>>(X, conv_w, conv_b, out_proj, out, M);
}